// EMADecomp_74122545594743
// MI455X (gfx1250) — compile-verified
//
#include <hip/hip_runtime.h>
#include <cmath>
#include <stdint.h>

// EMA decomposition: x_low = EMA(x, a1); res = x - x_low; x_mid = EMA(res, a2); x_high = res - x_mid
// B=16, T=2048, D=1024, fp32.  T split into C=16 chunks of L=128.
#define B_  16
#define T_  2048
#define D_  1024
#define C_  16
#define L_  128

typedef float v2f __attribute__((ext_vector_type(2)));
typedef float v8f __attribute__((ext_vector_type(8)));

__device__ __forceinline__ float powi_(float r, int e) {
  float p = 1.0f;
  for (int i = 0; i < e; ++i) p *= r;
  return p;
}

// CDNA5 async global->LDS copy (ASYNCcnt-tracked).  vdst = LDS byte address, vaddr = 64-bit global VA.
__device__ __forceinline__ void async_b128(uint32_t lds_off, const float* g) {
  asm volatile("global_load_async_to_lds_b128 %0, %1, off"
               :: "v"(lds_off), "v"(g) : "memory");
}

// -----------------------------------------------------------------------------
// K1: per-(b, chunk, 16-channel group) WMMA scan producing chunk-end states
//     (low, mid) with zero initial state (exact chunk-local contribution).
// One wave per 16-channel tile column; 8 sequential 16x16 time tiles per chunk.
// -----------------------------------------------------------------------------
__global__ void __launch_bounds__(128)
ema_k1_endstate(const float* __restrict__ x, float2* __restrict__ e,
                float a1, float r1, float a2, float r2)
{
  __shared__ __align__(16) float smem[4 * 1024];   // 4 waves * (2 x-tiles + res-tile), padded rows

  const int tid  = threadIdx.x;
  const int wave = tid >> 5;
  const int lane = tid & 31;
  const int hi   = lane >> 4;      // lane half (selects K/M sub-blocks per WMMA layout)
  const int n    = lane & 15;      // column (channel) for B/C operands
  const int m    = n;              // row for A operand (M = lane%16)

  const int g  = blockIdx.x * 4 + wave;   // 16-channel group 0..63
  const int c  = blockIdx.y;              // chunk
  const int b  = blockIdx.z;              // batch
  const int d0 = g * 16;

  const float* xbase = x + ((size_t)(b * T_) + (size_t)c * L_) * D_ + d0;

  float* mybase = smem + wave * 1024;                 // 320w buf0 | 320w buf1 | 320w res
  const uint32_t mybytes = (uint32_t)(uintptr_t)mybase;
  float* resbuf = mybase + 640;

  // Per-lane A-matrix fragments for the 4 chained 16x16x4 WMMAs (K blocks of 4).
  // A layout: VGPR0 holds A[m][4s+2*hi], VGPR1 holds A[m][4s+2*hi+1].
  float alow0[4], alow1[4], amid0[4], amid1[4];
  #pragma unroll
  for (int s = 0; s < 4; ++s) {
    int k0 = 4 * s + 2 * hi;
    int k1 = k0 + 1;
    alow0[s] = (k0 <= m) ? a1 * powi_(r1, m - k0) : 0.0f;
    alow1[s] = (k1 <= m) ? a1 * powi_(r1, m - k1) : 0.0f;
    amid0[s] = (k0 <= m) ? a2 * powi_(r2, m - k0) : 0.0f;
    amid1[s] = (k1 <= m) ? a2 * powi_(r2, m - k1) : 0.0f;
  }
  // Global t=0: first EMA sample passes through unscaled -> column 0 of A is r1^m.
  const float afirst0 = (c == 0 && hi == 0) ? powi_(r1, m) : alow0[0];

  // Carry decay coefficients for C rows (row M = r + 8*hi): r^(M+1)
  float clow[8], cmid[8];
  #pragma unroll
  for (int r = 0; r < 8; ++r) {
    clow[r] = powi_(r1, r + 8 * hi + 1);
    cmid[r] = powi_(r2, r + 8 * hi + 1);
  }

  const int arow = lane >> 2;        // 0..7 (tile row per async lane)
  const int acol = (lane & 3) * 4;   // channel sub-column (floats)

  auto issue = [&](int j, int buf) {
    #pragma unroll
    for (int h = 0; h < 2; ++h) {
      int row = h * 8 + arow;
      const float* gp = xbase + (size_t)(j * 16 + row) * D_ + acol;
      uint32_t lp = mybytes + (uint32_t)(buf * 1280 + row * 80 + acol * 4); // 20-word padded rows
      async_b128(lp, gp);
    }
  };

  issue(0, 0);
  float carrylow = 0.0f, carrymid = 0.0f;

  for (int j = 0; j < 8; ++j) {
    const int buf = j & 1;
    if (j < 7) issue(j + 1, buf ^ 1);
    if (j < 7) asm volatile("s_wait_asynccnt 0x2" ::: "memory");
    else       asm volatile("s_wait_asynccnt 0x0" ::: "memory");
    __builtin_amdgcn_wave_barrier();

    const float* xb = mybase + buf * 320;

    // ---- low EMA tile: D = A_low * X + r1^{m+1} (x) carry ----
    v8f accL;
    #pragma unroll
    for (int r = 0; r < 8; ++r) accL[r] = clow[r] * carrylow;
    #pragma unroll
    for (int s = 0; s < 4; ++s) {
      v2f av, bv;
      av[0] = (s == 0 && j == 0) ? afirst0 : alow0[s];
      av[1] = alow1[s];
      // B layout: VGPR0 -> K = 4s + 2*hi, VGPR1 -> K = 4s + 1 + 2*hi, col = lane%16
      bv[0] = xb[(4 * s     + 2 * hi) * 20 + n];
      bv[1] = xb[(4 * s + 1 + 2 * hi) * 20 + n];
      accL = __builtin_amdgcn_wmma_f32_16x16x4_f32(false, av, false, bv,
                                                   (short)0, accL, false, false);
    }

    // ---- residual tile (x - low), transposed back to time-major via LDS ----
    #pragma unroll
    for (int r = 0; r < 8; ++r) {
      int row = r + 8 * hi;                       // C layout: VGPR r holds row r + 8*hi
      resbuf[row * 20 + n] = xb[row * 20 + n] - accL[r];
    }
    __builtin_amdgcn_wave_barrier();

    // ---- mid EMA tile on residual ----
    v8f accM;
    #pragma unroll
    for (int r = 0; r < 8; ++r) accM[r] = cmid[r] * carrymid;
    #pragma unroll
    for (int s = 0; s < 4; ++s) {
      v2f av, bv;
      av[0] = amid0[s];
      av[1] = amid1[s];
      bv[0] = resbuf[(4 * s     + 2 * hi) * 20 + n];
      bv[1] = resbuf[(4 * s + 1 + 2 * hi) * 20 + n];
      accM = __builtin_amdgcn_wmma_f32_16x16x4_f32(false, av, false, bv,
                                                   (short)0, accM, false, false);
    }

    // tile-end states (row M=15 lives in VGPR7 of lanes 16..31) -> broadcast per column
    carrylow = __shfl(accL[7], 16 + n, 32);
    carrymid = __shfl(accM[7], 16 + n, 32);
  }

  if (hi == 0) {
    e[((size_t)(b * C_ + c)) * D_ + d0 + n] = make_float2(carrylow, carrymid);
  }
}

// -----------------------------------------------------------------------------
// K2: serial carry propagation across the 16 chunks (tiny).  Exact 2x2 transition
//     over L zero-input steps: low' = p1*low ; mid' = m21*low + p2*mid.
// -----------------------------------------------------------------------------
__global__ void ema_k2_carry(const float2* __restrict__ e, float2* __restrict__ cin,
                             float p1, float p2, float m21)
{
  int gid = blockIdx.x * blockDim.x + threadIdx.x;   // 0 .. B*D-1
  if (gid >= B_ * D_) return;
  int b = gid / D_;
  int d = gid - b * D_;
  float lo = 0.0f, mi = 0.0f;
  for (int c = 0; c < C_; ++c) {
    size_t idx = ((size_t)(b * C_ + c)) * D_ + d;
    cin[idx] = make_float2(lo, mi);                  // carry-in for chunk c (state at end of c-1)
    float2 ec = e[idx];
    float nlo = ec.x + p1 * lo;
    float nmi = ec.y + (m21 * lo + p2 * mi);
    lo = nlo; mi = nmi;
  }
}

// -----------------------------------------------------------------------------
// K3: final scan, one thread per (b, chunk, channel).  x re-read hits the 192MB
//     L2 (warmed by K1); all 384 MiB of outputs stored with non-temporal hints
//     so they do not evict x from L2.
// -----------------------------------------------------------------------------
__global__ void __launch_bounds__(256)
ema_k3_scan(const float* __restrict__ x, const float2* __restrict__ cin,
            float* __restrict__ out, float a1, float r1, float a2, float r2)
{
  const int d = blockIdx.x * blockDim.x + threadIdx.x;  // 0..D-1
  const int c = blockIdx.y;
  const int b = blockIdx.z;
  const size_t BTD = (size_t)B_ * T_ * D_;
  const size_t base = ((size_t)(b * T_) + (size_t)c * L_) * D_ + d;
  const float* xp = x + base;
  float* hq = out + base;             // x_high
  float* mq = out + BTD + base;       // x_mid
  float* lq = out + 2 * BTD + base;   // x_low

  float lo, mi;
  int j0;
  if (c == 0) {
    float x0 = xp[0];                 // t=0: low=x0, res=0, mid=0, high=0 (first sample unscaled)
    __builtin_nontemporal_store(0.0f, hq);
    __builtin_nontemporal_store(0.0f, mq);
    __builtin_nontemporal_store(x0,  lq);
    lo = x0; mi = 0.0f; j0 = 1;
  } else {
    float2 cc = cin[((size_t)(b * C_ + c)) * D_ + d];
    lo = cc.x; mi = cc.y; j0 = 0;
  }

  #pragma unroll 4
  for (int j = j0; j < L_; ++j) {
    size_t o = (size_t)j * D_;
    float xv = xp[o];
    lo = fmaf(a1, xv, r1 * lo);       // low_t  = a1*x + (1-a1)*low
    float res = xv - lo;
    mi = fmaf(a2, res, r2 * mi);      // mid_t  = a2*res + (1-a2)*mid
    float hv = res - mi;
    __builtin_nontemporal_store(hv, hq + o);
    __builtin_nontemporal_store(mi, mq + o);
    __builtin_nontemporal_store(lo, lq + o);
  }
}

extern "C" void kernel_launch(void* const* d_in, const int* in_sizes, int n_in,
                              void* d_out, int out_size, void* d_ws, size_t ws_size,
                              hipStream_t stream)
{
  (void)in_sizes; (void)n_in; (void)out_size; (void)ws_size;
  const float* x = (const float*)d_in[0];
  float* out = (float*)d_out;

  const double a1 = 0.005, a2 = 0.02;
  const double r1 = 1.0 - a1, r2 = 1.0 - a2;
  const double p1  = std::pow(r1, (double)L_);
  const double p2  = std::pow(r2, (double)L_);
  const double m21 = -a2 * r1 * (std::pow(r1, (double)L_) - std::pow(r2, (double)L_)) / (r1 - r2);

  // workspace: chunk-end states + carry-ins, B*C*D float2 each (2 MiB + 2 MiB)
  float2* e   = (float2*)d_ws;
  float2* cin = e + (size_t)B_ * C_ * D_;

  dim3 g1(D_ / 64, C_, B_);                 // 4096 blocks x 4 waves
  ema_k1_endstate<<<g1, 128, 0, stream>>>(x, e, (float)a1, (float)r1, (float)a2, (float)r2);

  ema_k2_carry<<<(B_ * D_ + 255) / 256, 256, 0, stream>>>(e, cin,
                                                          (float)p1, (float)p2, (float)m21);

  dim3 g3(D_ / 256, C_, B_);                // 1024 blocks x 8 waves
  ema_k3_scan<<<g3, 256, 0, stream>>>(x, cin, out, (float)a1, (float)r1, (float)a2, (float)r2);
}